// ACE_54881092108426
// MI455X (gfx1250) — compile-verified
//
#include <hip/hip_runtime.h>
#include <math.h>

#define N_NODES 20000
#define N_EDGES 640000
#define NSP 8
#define NMAX 8
#define NUM_BASIS 928   // 8*4 + 8*4*7*4
#define RCUT 5.0f
#define PI_F 3.14159265358979323846f

typedef __attribute__((ext_vector_type(2))) float v2f;
typedef __attribute__((ext_vector_type(8))) float v8f;

__global__ void ace_zero_kernel(float* __restrict__ p, int n) {
    int i = blockIdx.x * blockDim.x + threadIdx.x;
    if (i < n) p[i] = 0.0f;
}

// ---------------------------------------------------------------------------
// Edge kernel: bessel + spherical harmonics, scatter into A[recv][8][16]
// via L2-resident f32 atomics, plus pair-energy scatter.
// ---------------------------------------------------------------------------
__global__ __launch_bounds__(256) void ace_edge_kernel(
    const float* __restrict__ vec, const int* __restrict__ senders,
    const int* __restrict__ receivers, const int* __restrict__ species,
    const float* __restrict__ wA, const float* __restrict__ c_pair,
    float* __restrict__ Aacc, float* __restrict__ Epair) {
    int e = blockIdx.x * blockDim.x + threadIdx.x;
    if (e >= N_EDGES) return;

    float x = vec[3 * e + 0], y = vec[3 * e + 1], z = vec[3 * e + 2];
    float r = sqrtf(x * x + y * y + z * z + 1e-12f);
    float inv_r = 1.0f / r;
    float d = r * (1.0f / RCUT);

    // polynomial envelope, p = 6
    float env = 0.0f;
    if (d < 1.0f) {
        float d2 = d * d, d3 = d2 * d, d6 = d3 * d3;
        float d7 = d6 * d, d8 = d7 * d;
        env = 1.0f - 28.0f * d6 + 48.0f * d7 - 21.0f * d8;
    }

    // sin((n+1)*pi*d) via Chebyshev recurrence: 2 transcendentals, not 8
    float Rnl[NMAX];
    float base = 0.63245553203f /* sqrt(2/RCUT) */ * env * inv_r;
    float t  = PI_F * d;
    float sn = __sinf(t);
    float twoc = 2.0f * __cosf(t);
    float sp = 0.0f;
    Rnl[0] = base * sn;
#pragma unroll
    for (int n = 1; n < NMAX; n++) {
        float s2 = twoc * sn - sp;
        sp = sn;
        sn = s2;
        Rnl[n] = base * sn;
    }

    float ux = x * inv_r, uy = y * inv_r, uz = z * inv_r;
    float xx = ux * ux, yy = uy * uy, zz = uz * uz;
    float Y[16];
    Y[0]  = 0.28209479177387814f;
    Y[1]  = 0.4886025119029199f * uy;
    Y[2]  = 0.4886025119029199f * uz;
    Y[3]  = 0.4886025119029199f * ux;
    Y[4]  = 1.0925484305920792f * ux * uy;
    Y[5]  = 1.0925484305920792f * uy * uz;
    Y[6]  = 0.31539156525252005f * (3.0f * zz - 1.0f);
    Y[7]  = 1.0925484305920792f * ux * uz;
    Y[8]  = 0.5462742152960396f * (xx - yy);
    Y[9]  = 0.5900435899266435f * uy * (3.0f * xx - yy);
    Y[10] = 2.8906114426405543f * ux * uy * uz;
    Y[11] = 0.4570457994644658f * uy * (5.0f * zz - 1.0f);
    Y[12] = 0.3731763325901154f * uz * (5.0f * zz - 3.0f);
    Y[13] = 0.4570457994644658f * ux * (5.0f * zz - 1.0f);
    Y[14] = 1.445305722320977f * uz * (xx - yy);
    Y[15] = 0.5900435899266435f * ux * (xx - 3.0f * yy);

    int zs   = species[senders[e]];
    int recv = receivers[e];
    int zi   = species[recv];

    float coef[NMAX];
#pragma unroll
    for (int n = 0; n < NMAX; n++) coef[n] = wA[zs * NMAX + n] * Rnl[n];

    float* Anode = Aacc + (size_t)recv * 128;
#pragma unroll
    for (int n = 0; n < NMAX; n++) {
#pragma unroll
        for (int lm = 0; lm < 16; lm++)
            atomicAdd(&Anode[n * 16 + lm], coef[n] * Y[lm]);
    }

    const float* cp = c_pair + ((size_t)zi * NSP + zs) * NMAX;
    float ep = 0.0f;
#pragma unroll
    for (int n = 0; n < NMAX; n++) ep += cp[n] * Rnl[n];
    atomicAdd(&Epair[recv], 0.5f * ep);
}

// ---------------------------------------------------------------------------
// Node kernel: one wave32 per 16-node tile.
//  phase 1: P[16][4x8] power spectrum into LDS (stride 33 = bank-spread)
//  phase 2: G = B @ c_read^T via V_WMMA_F32_16X16X4_F32 (exact f32 GEMM,
//           M=16 nodes, N=16 (8 species + zero pad), K=928 in 232 chunks).
//           K-loop split: 8 chunks of B1 (direct P), 224 chunks of Q
//           driven by nested (a,l,k') counters -> branchless, EXEC stays 1s.
//  phase 3: per-node species select of G, E = G + E0 + pair, masked
//  phase 4: emit B rows (B1 direct, Q via branchless magic-div indexing)
// ---------------------------------------------------------------------------
__global__ __launch_bounds__(32) void ace_node_kernel(
    const int* __restrict__ species, const float* __restrict__ node_mask,
    const float* __restrict__ c_read, const float* __restrict__ E0,
    const float* __restrict__ Aacc, const float* __restrict__ Epair,
    float* __restrict__ outE, float* __restrict__ outB) {
    __shared__ float P_lds[16 * 33];
    __shared__ float G_lds[16 * 16];

    const int lane = threadIdx.x;
    const int n8   = lane & 15;   // node-in-tile (A frag) / species col (B frag)
    const int hh   = lane >> 4;   // half-wave selector -> K offset 2*hh
    const int tile = blockIdx.x;
    const int gnode = tile * 16 + n8;

    // ---- phase 1: each half-lane computes 4 of the 8 'a' channels of P ----
    {
        const float* An = Aacc + (size_t)gnode * 128 + hh * 64;
        float* Pn = &P_lds[n8 * 33 + hh * 16];
#pragma unroll
        for (int a = 0; a < 4; a++) {
            float4 f0 = *(const float4*)(An + a * 16 + 0);
            float4 f1 = *(const float4*)(An + a * 16 + 4);
            float4 f2 = *(const float4*)(An + a * 16 + 8);
            float4 f3 = *(const float4*)(An + a * 16 + 12);
            Pn[a * 4 + 0] = f0.x * f0.x;
            Pn[a * 4 + 1] = f0.y * f0.y + f0.z * f0.z + f0.w * f0.w;
            Pn[a * 4 + 2] = f1.x * f1.x + f1.y * f1.y + f1.z * f1.z +
                            f1.w * f1.w + f2.x * f2.x;
            Pn[a * 4 + 3] = f2.y * f2.y + f2.z * f2.z + f2.w * f2.w +
                            f3.x * f3.x + f3.y * f3.y + f3.z * f3.z +
                            f3.w * f3.w;
        }
    }
    __syncthreads();

    // ---- phase 2: WMMA f32 16x16x4 GEMM over K = 928 ----
    const float* Pmine = &P_lds[n8 * 33];
    const float* crow  = c_read + (size_t)(n8 & 7) * NUM_BASIS;
    const float  bmask = (n8 < NSP) ? 1.0f : 0.0f;  // zero-pad species 8..15
    const int    koff  = 2 * hh;                    // this lane's K sub-offset

    v8f acc = {};

    // chunks 0..7: b = 4c+koff < 32 -> B1 region, direct P reads
#pragma unroll
    for (int c = 0; c < 8; c++) {
        int kb = 4 * c + koff;
        v2f av, bv;
        av.x = Pmine[kb];
        av.y = Pmine[kb + 1];
        bv.x = crow[kb] * bmask;
        bv.y = crow[kb + 1] * bmask;
        acc = __builtin_amdgcn_wmma_f32_16x16x4_f32(
            false, av, false, bv, (short)0, acc, false, false);
    }

    // chunks 8..231: Q region, q = a*112 + l*28 + kp*4 + m, m = koff + j
    {
        int kb = 32 + koff;
        for (int a = 0; a < 8; a++) {
#pragma unroll
            for (int l = 0; l < 4; l++) {
                float Pal = Pmine[a * 4 + l];
#pragma unroll
                for (int kp = 0; kp < 7; kp++) {
                    int k = kp + (kp >= a ? 1 : 0);
                    v2f av, bv;
                    av.x = Pal * Pmine[k * 4 + koff];
                    av.y = Pal * Pmine[k * 4 + koff + 1];
                    bv.x = crow[kb] * bmask;
                    bv.y = crow[kb + 1] * bmask;
                    acc = __builtin_amdgcn_wmma_f32_16x16x4_f32(
                        false, av, false, bv, (short)0, acc, false, false);
                    kb += 4;
                }
            }
        }
    }

    // ---- phase 3: D layout (M = vgpr + 8*hh, N = n8) -> LDS -> select ----
#pragma unroll
    for (int v = 0; v < 8; v++)
        G_lds[(v + 8 * hh) * 16 + n8] = acc[v];
    __syncthreads();

    if (hh == 0) {
        int zi = species[gnode];
        float e = G_lds[n8 * 16 + zi] + E0[zi] + Epair[gnode];
        outE[gnode] = e * node_mask[gnode];
    }

    // ---- phase 4: emit B rows, coalesced across lanes, branchless ----
    for (int ni = 0; ni < 16; ni++) {
        const float* Pi = &P_lds[ni * 33];
        float* Brow = outB + (size_t)(tile * 16 + ni) * NUM_BASIS;
        // B1: b = lane in [0,32)
        Brow[lane] = Pi[lane];
        // Q: b = 32 + q, q = lane + 32*it, it = 0..27
        const unsigned m = lane & 3u;
#pragma unroll 4
        for (int it = 0; it < 28; it++) {
            unsigned q  = (unsigned)lane + 32u * it;
            unsigned cc = q >> 2;              // 28*a + 7*l + kp
            unsigned a  = cc / 28u;
            unsigned r1 = cc - a * 28u;
            unsigned l  = r1 / 7u;
            unsigned kp = r1 - l * 7u;
            unsigned k  = kp + (kp >= a ? 1u : 0u);
            Brow[32 + q] = Pi[a * 4u + l] * Pi[k * 4u + m];
        }
    }
}

extern "C" void kernel_launch(void* const* d_in, const int* in_sizes, int n_in,
                              void* d_out, int out_size, void* d_ws, size_t ws_size,
                              hipStream_t stream) {
    const float* vectors   = (const float*)d_in[0];
    const int*   senders   = (const int*)d_in[1];
    const int*   receivers = (const int*)d_in[2];
    const int*   species   = (const int*)d_in[3];
    const float* node_mask = (const float*)d_in[4];
    const float* wA        = (const float*)d_in[5];
    const float* c_read    = (const float*)d_in[6];
    const float* E0        = (const float*)d_in[7];
    const float* c_pair    = (const float*)d_in[8];

    float* Aacc  = (float*)d_ws;                      // 20000*128 f32
    float* Epair = Aacc + (size_t)N_NODES * 128;      // 20000 f32
    float* outE  = (float*)d_out;                     // 20000 f32
    float* outB  = outE + N_NODES;                    // 20000*928 f32

    int ztotal = N_NODES * 128 + N_NODES;
    ace_zero_kernel<<<(ztotal + 255) / 256, 256, 0, stream>>>(Aacc, ztotal);
    ace_edge_kernel<<<(N_EDGES + 255) / 256, 256, 0, stream>>>(
        vectors, senders, receivers, species, wA, c_pair, Aacc, Epair);
    ace_node_kernel<<<N_NODES / 16, 32, 0, stream>>>(
        species, node_mask, c_read, E0, Aacc, Epair, outE, outB);
}